// WhisperSelfAttentionStepTVM_47588237640165
// MI455X (gfx1250) — compile-verified
//
#include <hip/hip_runtime.h>
#include <hip/hip_bf16.h>

typedef __attribute__((ext_vector_type(2))) float v2f;
typedef __attribute__((ext_vector_type(8))) float v8f;

#define BSZ 128
#define HN  20
#define TN  448
#define EN  1280
#define DH  64
#define QK_SCALE 0.125f   // 64^-0.5

// ---------------------------------------------------------------------------
// Fp32 WMMA GEMM:  Out[M,N] = (A[M,K] @ W[N,K]^T + bias) * scale
// One wave computes one 16x16 tile via V_WMMA_F32_16X16X4_F32; 4 waves/block.
// ---------------------------------------------------------------------------
__global__ __launch_bounds__(128) void gemm_wmma_f32(
    const float* __restrict__ A, const float* __restrict__ W,
    const float* __restrict__ bias, float* __restrict__ Out,
    int M, int N, int K, float scale)
{
  const int lane = threadIdx.x & 31;
  const int wave = threadIdx.x >> 5;
  const int l    = lane & 15;     // row/col within tile
  const int half = lane >> 4;     // K sub-offset selector

  const int n0 = (blockIdx.x * 4 + wave) * 16;
  const int m0 = blockIdx.y * 16;
  if (n0 >= N || m0 >= M) return;           // wave-uniform

  const float* arow = A + (size_t)(m0 + l) * K + 2 * half;
  const float* wrow = W + (size_t)(n0 + l) * K + 2 * half;

  v8f acc = {0.f, 0.f, 0.f, 0.f, 0.f, 0.f, 0.f, 0.f};
  for (int k0 = 0; k0 < K; k0 += 4) {
    // A tile 16x4: lanes 0-15 hold K=k0..k0+1, lanes 16-31 hold K=k0+2..k0+3
    v2f a = *(const v2f*)(arow + k0);
    // B tile 4x16: B[k][n] = W[n][k]; v0 = rows K=0/2, v1 = rows K=1/3
    v2f b = *(const v2f*)(wrow + k0);
    acc = __builtin_amdgcn_wmma_f32_16x16x4_f32(
        false, a, false, b, (short)0, acc, false, false);
  }

  const int col = n0 + l;
  const float bb = bias ? bias[col] : 0.0f;
#pragma unroll
  for (int r = 0; r < 8; ++r) {
    const int row = m0 + r + 8 * half;      // C/D layout: vgpr r -> M=r / r+8
    Out[(size_t)row * N + col] = (acc[r] + bb) * scale;
  }
}

// ---------------------------------------------------------------------------
// Fused attention decode + sliding-window cache shift.
// One block per (b,h). 256 threads = 8 waves. Each wave strides over t.
// Lane owns 2 of the 64 head-dims (float2, coalesced 256B/wave/row).
// While K/V rows are in registers for the dot products, they are also written
// shifted into updated_k/updated_v -> past cache is read exactly once.
// ---------------------------------------------------------------------------
__global__ __launch_bounds__(256) void attn_shift_kernel(
    const float* __restrict__ q,       // [B, E] already scaled
    const float* __restrict__ knew,    // [B, E]
    const float* __restrict__ vnew,    // [B, E]
    const float* __restrict__ past_k,  // [B, H, T, Dh]
    const float* __restrict__ past_v,  // [B, H, T, Dh]
    const float* __restrict__ mask,    // [B, 1, 1, T]
    float* __restrict__ attn_out,      // [B, E]
    float* __restrict__ upd_k,         // [B, H, T, Dh]
    float* __restrict__ upd_v)         // [B, H, T, Dh]
{
  const int bh = blockIdx.x;
  const int b  = bh / HN;
  const int h  = bh % HN;

  __shared__ float qs[DH];
  __shared__ float sc[TN + 1];
  __shared__ float red[256];
  __shared__ float part[8][DH];

  const int tid  = threadIdx.x;
  const int wave = tid >> 5;
  const int lane = tid & 31;
  const int d    = lane * 2;

  if (tid < DH) qs[tid] = q[(size_t)b * EN + h * DH + tid];
  __syncthreads();

  const float* kbase  = past_k + (size_t)bh * TN * DH;
  const float* vbase  = past_v + (size_t)bh * TN * DH;
  float*       ukbase = upd_k  + (size_t)bh * TN * DH;
  float*       uvbase = upd_v  + (size_t)bh * TN * DH;
  const float* knrow  = knew + (size_t)b * EN + h * DH;
  const float* vnrow  = vnew + (size_t)b * EN + h * DH;

  const float qx = qs[d];
  const float qy = qs[d + 1];

  // ---- pass 1: scores over t = 0..448 (448 == new step) + K shift ----
  for (int t = wave; t <= TN; t += 8) {
    const float* krow = (t < TN) ? (kbase + (size_t)t * DH) : knrow;
    const float2 kv = *(const float2*)(krow + d);
    float p = kv.x * qx + kv.y * qy;
#pragma unroll
    for (int off = 16; off > 0; off >>= 1) p += __shfl_xor(p, off, 32);
    if (lane == 0) {
      const float m = (t < TN) ? mask[(size_t)b * TN + t] : 0.0f;
      sc[t] = p + m;
    }
    if (t >= 1)  // slot t goes to slot t-1; slot 0 is dropped
      *(float2*)(ukbase + (size_t)(t - 1) * DH + d) = kv;
  }
  __syncthreads();

  // ---- softmax: block max ----
  float lm = -1e30f;
  for (int t = tid; t <= TN; t += 256) lm = fmaxf(lm, sc[t]);
  red[tid] = lm;
  __syncthreads();
  for (int s = 128; s > 0; s >>= 1) {
    if (tid < s) red[tid] = fmaxf(red[tid], red[tid + s]);
    __syncthreads();
  }
  const float mx = red[0];
  __syncthreads();

  // ---- softmax: exp + block sum ----
  float ls = 0.0f;
  for (int t = tid; t <= TN; t += 256) {
    const float e = __expf(sc[t] - mx);
    sc[t] = e;
    ls += e;
  }
  red[tid] = ls;
  __syncthreads();
  for (int s = 128; s > 0; s >>= 1) {
    if (tid < s) red[tid] += red[tid + s];
    __syncthreads();
  }
  const float inv = 1.0f / red[0];

  // ---- pass 2: P @ V + V shift ----
  float ax = 0.0f, ay = 0.0f;
  for (int t = wave; t <= TN; t += 8) {
    const float* vrow = (t < TN) ? (vbase + (size_t)t * DH) : vnrow;
    const float2 vv = *(const float2*)(vrow + d);
    const float p = sc[t];
    ax += p * vv.x;
    ay += p * vv.y;
    if (t >= 1)
      *(float2*)(uvbase + (size_t)(t - 1) * DH + d) = vv;
  }
  part[wave][d]     = ax;
  part[wave][d + 1] = ay;
  __syncthreads();

  if (tid < DH) {
    float s = 0.0f;
#pragma unroll
    for (int i = 0; i < 8; ++i) s += part[i][tid];
    attn_out[(size_t)b * EN + h * DH + tid] = s * inv;
  }
}

// ---------------------------------------------------------------------------
extern "C" void kernel_launch(void* const* d_in, const int* in_sizes, int n_in,
                              void* d_out, int out_size, void* d_ws, size_t ws_size,
                              hipStream_t stream) {
  const float* hidden = (const float*)d_in[0];   // [128,1,1280]
  const float* past_k = (const float*)d_in[1];   // [128,20,448,64]
  const float* past_v = (const float*)d_in[2];
  const float* mask   = (const float*)d_in[3];   // [128,1,1,448]
  const float* q_w    = (const float*)d_in[4];
  const float* q_b    = (const float*)d_in[5];
  const float* k_w    = (const float*)d_in[6];
  const float* v_w    = (const float*)d_in[7];
  const float* v_b    = (const float*)d_in[8];
  const float* o_w    = (const float*)d_in[9];
  const float* o_b    = (const float*)d_in[10];

  float* out   = (float*)d_out;                          // [128,1,1280]
  float* upd_k = out + (size_t)BSZ * EN;                 // [128,20,448,64]
  float* upd_v = upd_k + (size_t)BSZ * HN * TN * DH;

  float* ws      = (float*)d_ws;
  float* q_ws    = ws;
  float* k_ws    = ws + 1 * (size_t)BSZ * EN;
  float* v_ws    = ws + 2 * (size_t)BSZ * EN;
  float* attn_ws = ws + 3 * (size_t)BSZ * EN;

  dim3 gblock(128);
  dim3 ggrid(EN / 64, BSZ / 16);  // 20 x 8; 4 wmma tiles per block

  // QKV projections (fp32 WMMA). q gets bias + 1/sqrt(Dh) scale fused.
  gemm_wmma_f32<<<ggrid, gblock, 0, stream>>>(hidden, q_w, q_b, q_ws,
                                              BSZ, EN, EN, QK_SCALE);
  gemm_wmma_f32<<<ggrid, gblock, 0, stream>>>(hidden, k_w, nullptr, k_ws,
                                              BSZ, EN, EN, 1.0f);
  gemm_wmma_f32<<<ggrid, gblock, 0, stream>>>(hidden, v_w, v_b, v_ws,
                                              BSZ, EN, EN, 1.0f);

  // Fused attention + cache shift: one block per (b,h).
  attn_shift_kernel<<<dim3(BSZ * HN), dim3(256), 0, stream>>>(
      q_ws, k_ws, v_ws, past_k, past_v, mask, attn_ws, upd_k, upd_v);

  // Output projection (fp32 WMMA) straight into d_out.
  gemm_wmma_f32<<<ggrid, gblock, 0, stream>>>(attn_ws, o_w, o_b, out,
                                              BSZ, EN, EN, 1.0f);
}